// ConsitencyLoss_81587198754830
// MI455X (gfx1250) — compile-verified
//
#include <hip/hip_runtime.h>
#include <hip/hip_bf16.h>

typedef __attribute__((ext_vector_type(2))) float v2f;
typedef __attribute__((ext_vector_type(8))) float v8f;

#define N_TOTAL   44236800u   // 64*3*480*480
#define PLANE     230400u     // 480*480
#define NPLANES   192         // 64*3
#define NBLOCKS   2048
#define TPB       256

__device__ __forceinline__ float fast_sigmoid(float x) {
    // sigmoid(x) = 1 / (1 + exp(-x)) = rcp(1 + 2^(-x * log2(e)))
    float e = __builtin_amdgcn_exp2f(x * -1.44269504088896340736f); // v_exp_f32
    return __builtin_amdgcn_rcpf(1.0f + e);                        // v_rcp_f32
}

__global__ __launch_bounds__(TPB) void consistency_partial_kernel(
        const float* __restrict__ in,
        const long long* __restrict__ seg,
        float* __restrict__ partials) {
    __shared__ float lmask[NPLANES];
    __shared__ float wsum[TPB / 32];

    // Precompute the [B,C] mask (192 entries) once per block.
    if (threadIdx.x < NPLANES) {
        int b = (int)threadIdx.x / 3;
        int c = (int)threadIdx.x - 3 * b;
        long long s = seg[b];
        lmask[threadIdx.x] = (s != 0 && (long long)c != s) ? 1.0f : 0.0f;
    }
    __syncthreads();

    const unsigned lane          = threadIdx.x & 31u;
    const unsigned waveInBlock   = threadIdx.x >> 5;
    const unsigned wavesPerBlock = TPB >> 5;
    const unsigned globalWave    = blockIdx.x * wavesPerBlock + waveInBlock;
    const unsigned numWaves      = gridDim.x * wavesPerBlock;

    const v2f onesA = {1.0f, 1.0f};       // A matrix = all ones (16x4 f32, 2 VGPRs)
    v8f c0 = {}, c1 = {}, c2 = {}, c3 = {};

    // Wave-uniform grid-stride loop: each wave consumes 256 contiguous floats
    // per iteration (lane-contiguous groups of 8). N_TOTAL % 256 == 0, so the
    // guard never diverges within a wave and EXEC stays all-ones for WMMA.
    for (unsigned base = globalWave * 256u; base < N_TOTAL; base += numWaves * 256u) {
        unsigned idx = base + lane * 8u;
        const float4* p = reinterpret_cast<const float4*>(in + idx);
        float4 a0 = p[0];                  // global_load_b128
        float4 a1 = p[1];                  // global_load_b128
        float m = lmask[idx / PLANE];      // 8 | 230400 -> one plane per group

        v2f b0 = { fast_sigmoid(a0.x) * m, fast_sigmoid(a0.y) * m };
        v2f b1 = { fast_sigmoid(a0.z) * m, fast_sigmoid(a0.w) * m };
        v2f b2 = { fast_sigmoid(a1.x) * m, fast_sigmoid(a1.y) * m };
        v2f b3 = { fast_sigmoid(a1.z) * m, fast_sigmoid(a1.w) * m };

        // D = 1s x B + C : every WMMA folds 64 wave-wide f32 values into C.
        // (Each value lands in all 16 rows -> divide grand total by 16 at end.)
        c0 = __builtin_amdgcn_wmma_f32_16x16x4_f32(false, onesA, false, b0, (short)0, c0, false, false);
        c1 = __builtin_amdgcn_wmma_f32_16x16x4_f32(false, onesA, false, b1, (short)0, c1, false, false);
        c2 = __builtin_amdgcn_wmma_f32_16x16x4_f32(false, onesA, false, b2, (short)0, c2, false, false);
        c3 = __builtin_amdgcn_wmma_f32_16x16x4_f32(false, onesA, false, b3, (short)0, c3, false, false);
    }

    // Collapse the 4 accumulator fragments: total = sum over lanes & components.
    v8f cs = c0 + c1 + c2 + c3;
    float t = 0.0f;
#pragma unroll
    for (int i = 0; i < 8; ++i) t += cs[i];

    // wave32 butterfly reduction
#pragma unroll
    for (int off = 16; off >= 1; off >>= 1) t += __shfl_xor(t, off, 32);

    if (lane == 0) wsum[waveInBlock] = t;
    __syncthreads();
    if (threadIdx.x == 0) {
        float s = 0.0f;
#pragma unroll
        for (int i = 0; i < (int)(TPB / 32); ++i) s += wsum[i];
        partials[blockIdx.x] = s;   // still scaled by 16x
    }
}

__global__ __launch_bounds__(TPB) void consistency_finalize_kernel(
        const float* __restrict__ partials, int nparts,
        const long long* __restrict__ seg,
        float* __restrict__ out) {
    __shared__ float sdata[TPB];
    __shared__ float scnt[TPB];

    float t = 0.0f;
    for (int i = (int)threadIdx.x; i < nparts; i += TPB) t += partials[i];

    float cnt = 0.0f;
    if (threadIdx.x < 64) cnt = (seg[threadIdx.x] != 0) ? 1.0f : 0.0f;

    sdata[threadIdx.x] = t;
    scnt[threadIdx.x]  = cnt;
    __syncthreads();

    for (int s = TPB / 2; s > 0; s >>= 1) {
        if ((int)threadIdx.x < s) {
            sdata[threadIdx.x] += sdata[threadIdx.x + s];
            scnt[threadIdx.x]  += scnt[threadIdx.x + s];
        }
        __syncthreads();
    }

    if (threadIdx.x == 0) {
        float total = sdata[0] * (1.0f / 16.0f);          // undo A=ones 16-row fanout
        float denom = scnt[0] * 691200.0f + 1.0f;         // PIXELS_PER_SAMPLE = 480*480*3
        out[0] = total / denom;
    }
}

extern "C" void kernel_launch(void* const* d_in, const int* in_sizes, int n_in,
                              void* d_out, int out_size, void* d_ws, size_t ws_size,
                              hipStream_t stream) {
    (void)in_sizes; (void)n_in; (void)out_size; (void)ws_size;
    const float*     in  = (const float*)d_in[0];
    const long long* seg = (const long long*)d_in[1];
    float* out      = (float*)d_out;
    float* partials = (float*)d_ws;   // NBLOCKS floats of scratch

    consistency_partial_kernel<<<NBLOCKS, TPB, 0, stream>>>(in, seg, partials);
    consistency_finalize_kernel<<<1, TPB, 0, stream>>>(partials, NBLOCKS, seg, out);
}